// TopologyNetwork_80659485818987
// MI455X (gfx1250) — compile-verified
//
#include <hip/hip_runtime.h>
#include <stdint.h>
#include <stddef.h>

// Problem constants (match reference): B=1024 batch, N=5000 nodes/level,
// K=16 fan-in, L=10 levels, LeakyReLU(0.1).
#define BB 1024
#define NN 5000
#define KK 16
#define LL 10
#define SLOPE 0.1f

// ---------------------------------------------------------------------------
// CDNA5 async global->LDS path (gfx1250). Guarded so the file always compiles;
// fallback is a plain guarded load loop.
// Builtin signature (from hipcc diagnostic): 
//   void __builtin_amdgcn_global_load_async_to_lds_b128(
//       int4vec addrspace(1)* gsrc, int4vec addrspace(3)* ldst,
//       constant int offset, constant int cpol);
// ---------------------------------------------------------------------------
#if defined(__HIP_DEVICE_COMPILE__) && \
    __has_builtin(__builtin_amdgcn_global_load_async_to_lds_b128) && \
    __has_builtin(__builtin_amdgcn_s_wait_asynccnt)
#define USE_ASYNC_LDS 1
typedef __attribute__((ext_vector_type(4))) int v4i;
typedef __attribute__((address_space(1))) v4i glb_v4i;
typedef __attribute__((address_space(3))) v4i lds_v4i;
// Generic->AS1 is value-identical (flat global address == global address).
__device__ inline glb_v4i* to_glb(const void* p) {
  return (glb_v4i*)(uintptr_t)p;
}
// Generic->AS3: the LDS aperture has zero low 32 bits, so the AS3
// representation of a generic LDS pointer is just its low 32 bits (ISA 10.2).
__device__ inline lds_v4i* to_lds(void* p) {
  return (lds_v4i*)(uint32_t)(uintptr_t)p;
}
#else
#define USE_ASYNC_LDS 0
#endif

// ---------------------------------------------------------------------------
// Tiled transpose: in[R][C] -> out[C][R].  32x32 tiles staged in LDS.
// Loads use the async tensor path (global_load_async_to_lds_b128) when the
// 16B vector is fully in range and 16B-aligned; stores are float4 when legal.
// ---------------------------------------------------------------------------
__global__ __launch_bounds__(256) void topo_transpose_kernel(
    const float* __restrict__ in, float* __restrict__ out, int R, int C)
{
  __shared__ __align__(16) float tile[32][36];  // 36: 16B-aligned rows, few bank conflicts

  const int t  = threadIdx.x;          // 0..255
  const int c0 = blockIdx.x << 5;      // tile col origin (C axis)
  const int r0 = blockIdx.y << 5;      // tile row origin (R axis)

  // ---- load phase: thread t loads 4 consecutive elements of one row ----
  const int r  = t >> 3;               // 0..31
  const int c4 = (t & 7) << 2;         // 0,4,...,28
  const int gr = r0 + r;
  const int gc = c0 + c4;
  const bool fast = (gr < R) && (gc + 3 < C) && ((C & 3) == 0);

#if USE_ASYNC_LDS
  if (fast) {
    const float* gp = in + (size_t)gr * C + gc;
    __builtin_amdgcn_global_load_async_to_lds_b128(
        to_glb(gp), to_lds(&tile[r][c4]), /*offset=*/0, /*cpol=*/0);
  } else
#endif
  {
#pragma unroll
    for (int j = 0; j < 4; ++j) {
      float v = 0.0f;
      if (gr < R && (gc + j) < C) v = in[(size_t)gr * C + gc + j];
      tile[r][c4 + j] = v;
    }
  }

#if USE_ASYNC_LDS
  __builtin_amdgcn_s_wait_asynccnt(0);
#endif
  __syncthreads();

  // ---- store phase: thread t writes 4 consecutive elements of one out-row ----
  const int cc  = t >> 3;              // column of tile -> output row (C axis)
  const int r4  = (t & 7) << 2;        // 4 source rows -> output cols (R axis)
  const int oc  = c0 + cc;
  const int orr = r0 + r4;
  if (oc < C) {
    float4 v = make_float4(tile[r4 + 0][cc], tile[r4 + 1][cc],
                           tile[r4 + 2][cc], tile[r4 + 3][cc]);
    if (((R & 3) == 0) && (orr + 3 < R)) {
      *(float4*)(out + (size_t)oc * R + orr) = v;
    } else {
      if (orr + 0 < R) out[(size_t)oc * R + orr + 0] = v.x;
      if (orr + 1 < R) out[(size_t)oc * R + orr + 1] = v.y;
      if (orr + 2 < R) out[(size_t)oc * R + orr + 2] = v.z;
      if (orr + 3 < R) out[(size_t)oc * R + orr + 3] = v.w;
    }
  }
}

// ---------------------------------------------------------------------------
// One DAG level in [N,B] layout.  Block = one node (blockIdx.x = n), 256
// threads x float4 = full batch of 1024.  Parameters are wave-uniform ->
// compiler emits s_load_b128 scalar loads; each of the 16 gathers is a fully
// coalesced global_load_b128 (512B per wave) hitting L2-resident activations.
// ---------------------------------------------------------------------------
__global__ __launch_bounds__(256) void topo_level_kernel(
    const float* __restrict__ prev,   // [N,B]
    const int*   __restrict__ src,    // [N,K] this level
    const float* __restrict__ wts,    // [N,K] this level
    const float* __restrict__ bias,   // [N]   this level
    float*       __restrict__ next)   // [N,B]
{
  const int n = blockIdx.x;
  const int t = threadIdx.x;          // 0..255

  // Prefetch next node's parameter lines (emits global_prefetch_b8).
  if (t == 0 && (n + 1) < NN) {
    __builtin_prefetch(src + (size_t)(n + 1) * KK, 0, 0);
    __builtin_prefetch(wts + (size_t)(n + 1) * KK, 0, 0);
  }

  const int4*   s4 = (const int4*)(src + (size_t)n * KK);    // 64B aligned
  const float4* w4 = (const float4*)(wts + (size_t)n * KK);
  const int4   i0 = s4[0], i1 = s4[1], i2 = s4[2], i3 = s4[3];
  const float4 wa = w4[0], wb = w4[1], wc = w4[2], wd = w4[3];
  const float  bn = bias[n];

  const int idx[KK] = {i0.x, i0.y, i0.z, i0.w,  i1.x, i1.y, i1.z, i1.w,
                       i2.x, i2.y, i2.z, i2.w,  i3.x, i3.y, i3.z, i3.w};
  const float wk[KK] = {wa.x, wa.y, wa.z, wa.w,  wb.x, wb.y, wb.z, wb.w,
                        wc.x, wc.y, wc.z, wc.w,  wd.x, wd.y, wd.z, wd.w};

  float4 acc = make_float4(bn, bn, bn, bn);
#pragma unroll
  for (int k = 0; k < KK; ++k) {
    const float4 g = ((const float4*)(prev + (size_t)idx[k] * BB))[t];
    acc.x = fmaf(g.x, wk[k], acc.x);
    acc.y = fmaf(g.y, wk[k], acc.y);
    acc.z = fmaf(g.z, wk[k], acc.z);
    acc.w = fmaf(g.w, wk[k], acc.w);
  }

  // LeakyReLU(0.1) == max(x, 0.1*x) for slope < 1.
  float4 o;
  o.x = fmaxf(acc.x, SLOPE * acc.x);
  o.y = fmaxf(acc.y, SLOPE * acc.y);
  o.z = fmaxf(acc.z, SLOPE * acc.z);
  o.w = fmaxf(acc.w, SLOPE * acc.w);

  ((float4*)(next + (size_t)n * BB))[t] = o;
}

// ---------------------------------------------------------------------------
// Host orchestration.
//   bufW (ws)            : holds act after odd levels   -> level 9 result here
//   bufD (ws2nd or d_out): holds act0 and even levels
// Final: transpose bufW [N,B] -> d_out [B,N].  No read/write aliasing anywhere.
// ---------------------------------------------------------------------------
extern "C" void kernel_launch(void* const* d_in, const int* in_sizes, int n_in,
                              void* d_out, int out_size, void* d_ws, size_t ws_size,
                              hipStream_t stream) {
  (void)in_sizes; (void)n_in; (void)out_size;

  const float* x    = (const float*)d_in[0];   // [B,N]
  const int*   src  = (const int*)  d_in[1];   // [L-1,N,K]
  const float* wts  = (const float*)d_in[2];   // [L-1,N,K]
  const float* bias = (const float*)d_in[3];   // [L-1,N]
  float*       out  = (float*)d_out;           // [B,N]

  const size_t NB = (size_t)NN * BB;
  float* bufW = (float*)d_ws;
  float* bufD = (ws_size >= 2 * NB * sizeof(float)) ? (bufW + NB) : out;

  dim3 tb(256);

  // act0 = x^T  : [B,N] -> [N,B]
  dim3 tg_in((NN + 31) / 32, (BB + 31) / 32);
  topo_transpose_kernel<<<tg_in, tb, 0, stream>>>(x, bufD, BB, NN);

  const float* prev = bufD;
  for (int l = 1; l < LL; ++l) {
    float* dst = (l & 1) ? bufW : bufD;       // level 9 (odd) -> bufW (ws)
    const size_t po = (size_t)(l - 1) * NN;
    topo_level_kernel<<<NN, tb, 0, stream>>>(prev,
                                             src  + po * KK,
                                             wts  + po * KK,
                                             bias + po,
                                             dst);
    prev = dst;
  }

  // out = act9^T : [N,B] -> [B,N]
  dim3 tg_out((BB + 31) / 32, (NN + 31) / 32);
  topo_transpose_kernel<<<tg_out, tb, 0, stream>>>(prev, out, NN, BB);
}